// DensityAwareChamferDistance_26027501814344
// MI455X (gfx1250) — compile-verified
//
#include <hip/hip_runtime.h>

typedef __attribute__((ext_vector_type(2))) float v2f;
typedef __attribute__((ext_vector_type(8))) float v8f;

#define DACD_WAVES   8   // waves per block (256 threads)
#define DACD_CSPLIT  4   // column splits (blockIdx.y)

// Raw v_min_num_f32: skips LLVM's canonicalize (v_max x,x) scaffolding.
__device__ __forceinline__ float dacd_min_raw(float a, float b) {
    float r;
    asm("v_min_num_f32 %0, %1, %2" : "=v"(r) : "v"(a), "v"(b));
    return r;
}

// XOR lane swizzle within wave32 via ds_swizzle_b32 (imm = xor<<10 | and=0x1f).
#define DACD_SWZ_MIN(v, imm)                                                  \
    (v) = dacd_min_raw(                                                       \
        (v), __int_as_float(__builtin_amdgcn_ds_swizzle(__float_as_int(v),    \
                                                        (imm))))

// One 16-column tile: B cols from pre-negated packed float4 (bn), two WMMAs
// (row-tiles 0/1), running elementwise min. Named vars only -> stays in VGPRs.
#define DACD_TILE(bp)                                                         \
    do {                                                                      \
        v2f b;                                                                \
        b.x = lo ? (bp).x : (bp).z; /* K0: -2x | K2: -2z */                   \
        b.y = lo ? (bp).y : (bp).w; /* K1: -2y | K3: g2  */                   \
        v8f d0 = __builtin_amdgcn_wmma_f32_16x16x4_f32(                       \
            false, a0, false, b, (short)0, c0, false, false);                 \
        v8f d1 = __builtin_amdgcn_wmma_f32_16x16x4_f32(                       \
            false, a1, false, b, (short)0, c1, false, false);                 \
        _Pragma("unroll") for (int r = 0; r < 8; ++r) {                       \
            m0[r] = dacd_min_raw(m0[r], d0[r]);                               \
            m1[r] = dacd_min_raw(m1[r], d1[r]);                               \
        }                                                                     \
    } while (0)

// ---------------------------------------------------------------------------
// Kernel 0: init min arrays to +FLT_MAX bit pattern (for atomic-min combine).
// ---------------------------------------------------------------------------
__global__ void dacd_init_kernel(unsigned int* __restrict__ p, int n) {
    int i = blockIdx.x * blockDim.x + threadIdx.x;
    if (i < n) p[i] = 0x7F7FFFFFu;  // FLT_MAX
}

// ---------------------------------------------------------------------------
// Kernel 1: pack points. A-side: (x, y, z, n2). B-side: (-2x, -2y, -2z, n2).
// ---------------------------------------------------------------------------
__global__ void dacd_prep_kernel(const float* __restrict__ pts,
                                 float4* __restrict__ packedA,
                                 float4* __restrict__ packedB, int n) {
    int i = blockIdx.x * blockDim.x + threadIdx.x;
    if (i < n) {
        float x = pts[3 * i + 0];
        float y = pts[3 * i + 1];
        float z = pts[3 * i + 2];
        float n2 = x * x + y * y + z * z;
        packedA[i] = make_float4(x, y, z, n2);
        packedB[i] = make_float4(-2.f * x, -2.f * y, -2.f * z, n2);
    }
}

// ---------------------------------------------------------------------------
// Kernel 2: for each "row" point, min over a column-slice of squared dists.
// One wave owns two 16-row tiles; blockIdx.y selects a column slice.
// d2 tile = one V_WMMA_F32_16X16X4_F32:
//   A row i = (x_i, y_i, z_i, 1)      [lanes 0-15: K0,K1 ; lanes 16-31: K2,K3]
//   B col j = (-2x_j, -2y_j, -2z_j, g2_j)   (pre-negated in prep)
//   C[i][j] = p2_i  (row-constant, loop-invariant accumulator input)
// Column loop: chunks of 4 tiles, register double-buffered (named vars only,
// no allocas -> no LDS demotion). Partial row-mins clamped >=0 and combined
// with global_atomic_min_u32 (uint order == float order for non-negatives).
// ---------------------------------------------------------------------------
__global__ __launch_bounds__(256) void dacd_min_kernel(
    const float4* __restrict__ rows4,   // (x,y,z,norm2) for row points
    const float4* __restrict__ colsB,   // (-2x,-2y,-2z,norm2) for col points
    unsigned int* __restrict__ out_min, // per-row min (float bits), pre-init
    int tiles_per_split) {
    const int lane = threadIdx.x & 31;
    const int wave = threadIdx.x >> 5;
    const bool lo  = lane < 16;
    const int l16  = lane & 15;
    const int wrow = (blockIdx.x * DACD_WAVES + wave) * 32;

    // Row-tile operands (named, register-resident).
    v2f a0, a1;
    v8f c0, c1, m0, m1;
    {
        float4 ap0 = rows4[wrow + l16];
        float4 ap1 = rows4[wrow + 16 + l16];
        a0.x = lo ? ap0.x : ap0.z;
        a0.y = lo ? ap0.y : 1.0f;
        a1.x = lo ? ap1.x : ap1.z;
        a1.y = lo ? ap1.y : 1.0f;
#pragma unroll
        for (int r = 0; r < 8; ++r) {
            c0[r] = rows4[wrow + r + (lo ? 0 : 8)].w;
            c1[r] = rows4[wrow + 16 + r + (lo ? 0 : 8)].w;
            m0[r] = 3.4028235e38f;
            m1[r] = 3.4028235e38f;
        }
    }

    const float4* colp = colsB + (size_t)(blockIdx.y * tiles_per_split) * 16;
    const int n_chunks = tiles_per_split / 4;

    // Register double-buffer: 4 column tiles in flight.
    float4 q0 = colp[l16];
    float4 q1 = colp[16 + l16];
    float4 q2 = colp[32 + l16];
    float4 q3 = colp[48 + l16];

    for (int ch = 0; ch < n_chunks; ++ch) {
        const int pofs = ((ch + 1 < n_chunks) ? (ch + 1) : ch) * 64;
        float4 n0 = colp[pofs + l16];
        float4 n1 = colp[pofs + 16 + l16];
        float4 n2 = colp[pofs + 32 + l16];
        float4 n3 = colp[pofs + 48 + l16];

        DACD_TILE(q0);
        DACD_TILE(q1);
        DACD_TILE(q2);
        DACD_TILE(q3);

        q0 = n0; q1 = n1; q2 = n2; q3 = n3;
    }

    // Row-min: reduce across the 16 lanes of each half (rows r / r+8),
    // then clamp at 0 and atomically combine across column splits.
#pragma unroll
    for (int r = 0; r < 8; ++r) {
        float v0 = m0[r];
        float v1 = m1[r];
        DACD_SWZ_MIN(v0, 0x041F);  DACD_SWZ_MIN(v1, 0x041F);   // xor 1
        DACD_SWZ_MIN(v0, 0x081F);  DACD_SWZ_MIN(v1, 0x081F);   // xor 2
        DACD_SWZ_MIN(v0, 0x101F);  DACD_SWZ_MIN(v1, 0x101F);   // xor 4
        DACD_SWZ_MIN(v0, 0x201F);  DACD_SWZ_MIN(v1, 0x201F);   // xor 8
        if (l16 == 0) {
            int row = wrow + r + (lo ? 0 : 8);
            atomicMin(&out_min[row], __float_as_uint(fmaxf(v0, 0.0f)));
            atomicMin(&out_min[row + 16], __float_as_uint(fmaxf(v1, 0.0f)));
        }
    }
}

// ---------------------------------------------------------------------------
// Kernel 3: weighted normalization -> scalar output (min arrays hold float
// bits; all values are non-negative floats so plain reinterpretation works).
// ---------------------------------------------------------------------------
__global__ __launch_bounds__(256) void dacd_finalize_kernel(
    const float* __restrict__ minP, const float* __restrict__ wP, int P,
    const float* __restrict__ minG, const float* __restrict__ wG, int G,
    float* __restrict__ out) {
    __shared__ float sm[4 * 256];
    float swp = 0.f, swpm = 0.f, swg = 0.f, swgm = 0.f;
    for (int i = threadIdx.x; i < P; i += 256) {
        float w = wP[i];
        swp += w;
        swpm += w * minP[i];
    }
    for (int i = threadIdx.x; i < G; i += 256) {
        float w = wG[i];
        swg += w;
        swgm += w * minG[i];
    }
    sm[threadIdx.x]       = swp;
    sm[256 + threadIdx.x] = swpm;
    sm[512 + threadIdx.x] = swg;
    sm[768 + threadIdx.x] = swgm;
    __syncthreads();
    for (int s = 128; s > 0; s >>= 1) {
        if ((int)threadIdx.x < s) {
            sm[threadIdx.x]       += sm[threadIdx.x + s];
            sm[256 + threadIdx.x] += sm[256 + threadIdx.x + s];
            sm[512 + threadIdx.x] += sm[512 + threadIdx.x + s];
            sm[768 + threadIdx.x] += sm[768 + threadIdx.x + s];
        }
        __syncthreads();
    }
    if (threadIdx.x == 0) {
        out[0] = sm[256] / fmaxf(sm[0], 1e-9f) +
                 sm[768] / fmaxf(sm[512], 1e-9f);
    }
}

// ---------------------------------------------------------------------------
// Launcher
// ---------------------------------------------------------------------------
extern "C" void kernel_launch(void* const* d_in, const int* in_sizes, int n_in,
                              void* d_out, int out_size, void* d_ws, size_t ws_size,
                              hipStream_t stream) {
    const float* pred = (const float*)d_in[0];
    const float* gt   = (const float*)d_in[1];
    const float* wP   = (const float*)d_in[2];
    const float* wG   = (const float*)d_in[3];
    float* out        = (float*)d_out;

    const int P = in_sizes[0] / 3;
    const int G = in_sizes[1] / 3;

    // Workspace layout
    char* ws = (char*)d_ws;
    float4* SpA = (float4*)ws;                              // P*16 B
    float4* SgA = SpA + P;                                  // G*16 B
    float4* SpB = SgA + G;                                  // P*16 B
    float4* SgB = SpB + P;                                  // G*16 B
    unsigned int* minP = (unsigned int*)(SgB + G);          // P*4 B
    unsigned int* minG = minP + P;                          // G*4 B

    // Init min arrays to +FLT_MAX bits; pack both A-side and B-side formats.
    dacd_init_kernel<<<(P + G + 255) / 256, 256, 0, stream>>>(minP, P + G);
    dacd_prep_kernel<<<(P + 255) / 256, 256, 0, stream>>>(pred, SpA, SpB, P);
    dacd_prep_kernel<<<(G + 255) / 256, 256, 0, stream>>>(gt, SgA, SgB, G);

    // Pass 1: for each pred point, min over gt (columns split 4 ways)
    {
        dim3 grid(P / (32 * DACD_WAVES), DACD_CSPLIT);
        dacd_min_kernel<<<grid, 256, 0, stream>>>(SpA, SgB, minP,
                                                  (G / 16) / DACD_CSPLIT);
    }
    // Pass 2: for each gt point, min over pred
    {
        dim3 grid(G / (32 * DACD_WAVES), DACD_CSPLIT);
        dacd_min_kernel<<<grid, 256, 0, stream>>>(SgA, SpB, minG,
                                                  (P / 16) / DACD_CSPLIT);
    }

    // Weighted normalization
    dacd_finalize_kernel<<<1, 256, 0, stream>>>((const float*)minP, wP, P,
                                                (const float*)minG, wG, G, out);
}